// EDMultiheadRetention_38585986187935
// MI455X (gfx1250) — compile-verified
//
#include <hip/hip_runtime.h>
#include <hip/hip_bf16.h>

typedef float v2f __attribute__((ext_vector_type(2)));
typedef float v8f __attribute__((ext_vector_type(8)));
typedef int   v4i __attribute__((ext_vector_type(4)));
typedef __attribute__((address_space(1))) v4i* gptr_v4i;  // global
typedef __attribute__((address_space(3))) v4i* lptr_v4i;  // LDS

#define SEQ     128
#define DMODEL  4096
#define NHEADS  32
#define HDIM    128
#define LS      132                    // padded LDS row stride: 132 % 64 == 4 -> conflict-free strided reads
#define NKV_OFF (SEQ * DMODEL)         // 524288: new_kvs offset inside d_out
#define SEGSZ   (NHEADS * SEQ * HDIM)  // 524288 floats per q/k/v segment in workspace
#define SMEM_BYTES (4 * SEQ * LS * 4)  // q, k(->attn), v, kvs' : 270336 B

__device__ __forceinline__ v8f wmma_f32_k4(v2f a, v2f b, v8f c) {
  // D = A(16x4 f32) * B(4x16 f32) + C(16x16 f32)
  return __builtin_amdgcn_wmma_f32_16x16x4_f32(false, a, false, b, (short)0, c, false, false);
}

__device__ __forceinline__ float gelu_tanh(float x) {
  float x3 = x * x * x;
  return 0.5f * x * (1.0f + tanhf(0.7978845608028654f * (x + 0.044715f * x3)));
}

// ---------------------------------------------------------------------------
// Kernel 1: qkv = x @ W_qkv^T  (128 x 12288), pure f32 WMMA GEMM, no LDS.
// 192 blocks, each owns a 128-row x 64-col panel (W streamed once from HBM;
// x re-reads hit the 192MB L2). SGPR base + 32-bit lane offsets -> SADDR-form
// loads; unroll-4 folds k-stepping into load immediates.
// ---------------------------------------------------------------------------
__global__ __launch_bounds__(256) void qkv_gemm_kernel(
    const float* __restrict__ x,      // (128, 4096)
    const float* __restrict__ Wqkv,   // (12288, 4096)
    float* __restrict__ qkv)          // ws: 3 * 524288 floats
{
  const int tid  = threadIdx.x;
  const int lane = tid & 31;
  const int w    = tid >> 5;          // wave id 0..7 -> row strip
  const int lm   = lane & 15;
  const int lhi  = lane >> 4;
  const int m0   = w * 16;
  const int c0   = blockIdx.x * 64;   // global output-column base

  const unsigned aoff = (unsigned)((m0 + lm) * DMODEL + 2 * lhi);
  unsigned boff[4];
  #pragma unroll
  for (int ct = 0; ct < 4; ++ct)
    boff[ct] = (unsigned)(c0 + ct * 16 + lm) * DMODEL + 2 * lhi;

  v8f c[4];
  #pragma unroll
  for (int ct = 0; ct < 4; ++ct) c[ct] = (v8f){};

  #pragma unroll 4
  for (int kk = 0; kk < DMODEL; kk += 4) {
    v2f a = *(const v2f*)(x + aoff + kk);       // one A fetch feeds 4 WMMAs
    #pragma unroll
    for (int ct = 0; ct < 4; ++ct) {
      v2f b = *(const v2f*)(Wqkv + boff[ct] + kk);
      c[ct] = wmma_f32_k4(a, b, c[ct]);
    }
  }

  #pragma unroll
  for (int ct = 0; ct < 4; ++ct) {
    int cg   = c0 + ct * 16 + lm;              // global column
    int seg  = cg >> 12;                       // 0=q 1=k 2=v
    int rest = cg & 4095;
    int h    = rest >> 7;
    int j    = rest & 127;
    size_t base = (size_t)seg * SEGSZ + (size_t)h * (SEQ * HDIM) + (size_t)j;
    #pragma unroll
    for (int r = 0; r < 8; ++r) {
      int Mg = m0 + r + 8 * lhi;               // seq position
      qkv[base + (size_t)Mg * HDIM] = c[ct][r];
    }
  }
}

// ---------------------------------------------------------------------------
// Kernel 2: per-head retention. One block per head; q/k/v (L2-hot from ws)
// pulled into LDS (V via async global->LDS copy), then kvs update + causal
// attn + output, all on f32 WMMA.
// ---------------------------------------------------------------------------
__global__ __launch_bounds__(256) void retention_kernel(
    const float* __restrict__ qkv,    // ws from kernel 1
    const float* __restrict__ alpha,  // (32, 128)
    const float* __restrict__ kvs,    // (32, 128, 128)
    float* __restrict__ out)          // [0,524288): out ; [524288,1048576): new_kvs
{
  extern __shared__ float smem[];
  float* sQ  = smem;
  float* sK  = smem + 1 * SEQ * LS;   // reused as attn after phase D
  float* sV  = smem + 2 * SEQ * LS;
  float* sKV = smem + 3 * SEQ * LS;   // kvs * alpha^S

  const int h    = blockIdx.x;
  const int tid  = threadIdx.x;
  const int lane = tid & 31;
  const int w    = tid >> 5;
  const int lm   = lane & 15;
  const int lhi  = lane >> 4;
  const int m0   = w * 16;

  const float inv_sqrt_d = 1.0f / 64.0f;       // 1/sqrt(4096)
  const float* alphaH = alpha + (size_t)h * HDIM;
  const float* wsQ  = qkv + 0 * SEGSZ + (size_t)h * (SEQ * HDIM);
  const float* wsK  = qkv + 1 * SEGSZ + (size_t)h * (SEQ * HDIM);
  const float* wsV  = qkv + 2 * SEGSZ + (size_t)h * (SEQ * HDIM);
  const float* kvsH = kvs + (size_t)h * (HDIM * HDIM);

  // ---- V: raw copy via CDNA5 async global->LDS (ASYNCcnt-tracked) ----
  bool async_v = false;
#if defined(__has_builtin)
#if __has_builtin(__builtin_amdgcn_global_load_async_to_lds_b128)
  async_v = true;
  for (int q4 = tid; q4 < SEQ * HDIM / 4; q4 += 256) {
    int s = q4 >> 5, jq = q4 & 31;             // row s, 16B chunk jq within row
    __builtin_amdgcn_global_load_async_to_lds_b128(
        (gptr_v4i)(wsV + (size_t)q4 * 4),
        (lptr_v4i)(sV + s * LS + jq * 4),
        0, 0);
  }
#endif
#endif

  // ---- Fill LDS: decay-scaled q/k, kvs*alpha^S (and V if no async path) ----
  for (int idx = tid; idx < SEQ * HDIM; idx += 256) {
    int s = idx >> 7, j = idx & 127;
    float la = __logf(alphaH[j]);
    sQ[s * LS + j] = wsQ[idx] * __expf((float)(s - (SEQ - 1)) * la) * inv_sqrt_d;
    sK[s * LS + j] = wsK[idx] * __expf((float)((SEQ - 1) - s) * la) * inv_sqrt_d;
    if (!async_v) sV[s * LS + j] = wsV[idx];
    sKV[s * LS + j] = kvsH[idx] * __expf(128.0f * __logf(alphaH[s]));
  }
  if (async_v) {
#if defined(__has_builtin)
#if __has_builtin(__builtin_amdgcn_s_wait_asynccnt)
    __builtin_amdgcn_s_wait_asynccnt(0);
#else
    asm volatile("s_wait_asynccnt 0x0" ::: "memory");
#endif
#endif
  }
  __syncthreads();

  // ---- Phase C: new_kvs = kvs*alpha^S + k^T v  -> global ----
  for (int jt = 0; jt < 8; ++jt) {
    v8f c = {};
    #pragma unroll
    for (int kk = 0; kk < SEQ; kk += 4) {       // contract over seq
      v2f a, b;
      a.x = sK[(kk + 2 * lhi)     * LS + m0 + lm];     // A = k^T
      a.y = sK[(kk + 2 * lhi + 1) * LS + m0 + lm];
      b.x = sV[(kk + 2 * lhi)     * LS + jt * 16 + lm];
      b.y = sV[(kk + 2 * lhi + 1) * LS + jt * 16 + lm];
      c = wmma_f32_k4(a, b, c);
    }
    #pragma unroll
    for (int r = 0; r < 8; ++r) {
      int Mg = m0 + r + 8 * lhi;
      int Ng = jt * 16 + lm;
      out[NKV_OFF + (size_t)h * (HDIM * HDIM) + Mg * HDIM + Ng] = c[r] + sKV[Mg * LS + Ng];
    }
  }

  // ---- Phase D: attn = tril(q k^T), row strip w, tiles 0..w in regs ----
  v8f att[8];
  #pragma unroll
  for (int jt = 0; jt < 8; ++jt) {
    v8f c = {};
    if (jt <= w) {                              // wave-uniform: EXEC stays all-ones
      #pragma unroll
      for (int kk = 0; kk < HDIM; kk += 4) {
        v2f a = *(const v2f*)&sQ[(m0 + lm)      * LS + kk + 2 * lhi];
        v2f b = *(const v2f*)&sK[(jt * 16 + lm) * LS + kk + 2 * lhi];  // B = k^T
        c = wmma_f32_k4(a, b, c);
      }
      if (jt == w) {                            // diagonal tile: causal mask
        #pragma unroll
        for (int r = 0; r < 8; ++r)
          if ((jt * 16 + lm) > (m0 + r + 8 * lhi)) c[r] = 0.0f;
      }
    }
    att[jt] = c;
  }
  __syncthreads();                              // all reads of sK done

  #pragma unroll
  for (int jt = 0; jt < 8; ++jt) {              // store attn into sK slot
    if (jt <= w) {
      #pragma unroll
      for (int r = 0; r < 8; ++r)
        sK[(m0 + r + 8 * lhi) * LS + jt * 16 + lm] = att[jt][r];
    }
  }
  __syncthreads();

  // ---- Phase E: out = gelu(attn @ v + q @ (kvs*alpha^S)) ----
  const float* sATT = sK;
  const int kmax = 16 * (w + 1);                // attn strip w is zero beyond diag block
  for (int nt = 0; nt < 8; ++nt) {
    v8f c = {};
    for (int kk = 0; kk < kmax; kk += 4) {      // attn @ v
      v2f a = *(const v2f*)&sATT[(m0 + lm) * LS + kk + 2 * lhi];
      v2f b;
      b.x = sV[(kk + 2 * lhi)     * LS + nt * 16 + lm];
      b.y = sV[(kk + 2 * lhi + 1) * LS + nt * 16 + lm];
      c = wmma_f32_k4(a, b, c);
    }
    #pragma unroll
    for (int kk = 0; kk < HDIM; kk += 4) {      // q @ kvs'
      v2f a = *(const v2f*)&sQ[(m0 + lm) * LS + kk + 2 * lhi];
      v2f b;
      b.x = sKV[(kk + 2 * lhi)     * LS + nt * 16 + lm];
      b.y = sKV[(kk + 2 * lhi + 1) * LS + nt * 16 + lm];
      c = wmma_f32_k4(a, b, c);
    }
    #pragma unroll
    for (int r = 0; r < 8; ++r) {
      int Mg = m0 + r + 8 * lhi;                // seq position
      int Ng = nt * 16 + lm;                    // head channel
      out[(size_t)Mg * DMODEL + (size_t)h * HDIM + Ng] = gelu_tanh(c[r]);
    }
  }
}

extern "C" void kernel_launch(void* const* d_in, const int* in_sizes, int n_in,
                              void* d_out, int out_size, void* d_ws, size_t ws_size,
                              hipStream_t stream) {
  (void)in_sizes; (void)n_in; (void)ws_size; (void)out_size;
  const float* x     = (const float*)d_in[0];
  const float* Wqkv  = (const float*)d_in[1];
  const float* alpha = (const float*)d_in[2];
  const float* kvs   = (const float*)d_in[3];
  float* out = (float*)d_out;
  float* ws  = (float*)d_ws;                    // needs 3*524288*4 = 6 MB

  // allow >64KB dynamic LDS (WGP has 320KB); idempotent, not a stream op
  (void)hipFuncSetAttribute((const void*)retention_kernel,
                            hipFuncAttributeMaxDynamicSharedMemorySize, SMEM_BYTES);

  qkv_gemm_kernel<<<3 * DMODEL / 64, 256, 0, stream>>>(x, Wqkv, ws);
  retention_kernel<<<NHEADS, 256, SMEM_BYTES, stream>>>(ws, alpha, kvs, out);
}